// NEG_loss_32581621908004
// MI455X (gfx1250) — compile-verified
//
#include <hip/hip_runtime.h>

typedef __attribute__((ext_vector_type(2))) float v2f;
typedef __attribute__((ext_vector_type(8))) float v8f;

#define B_ 32
#define S_ 256
#define D_ 128
#define NNEG 15   // K-1 negatives per position

// One wave computes, for one (batch b, 32-row i-tile, target position j):
//   D[32 x 16] = latent_tile(32x128) x candidates_j(128x16)
// as two independent 16x16 accumulator chains sharing every B fragment
// (1 b64 load feeds 2 v_wmma_f32_16x16x4_f32). Then logsumexp over the 16
// candidate columns (half-wave shuffle reduction) and
// out[b,i,j] = D[i,15] - lse  for 32 i's.
__global__ __launch_bounds__(256)
void neg_loss_kernel(const float* __restrict__ latent,   // [B,S,D]
                     const int*   __restrict__ labels,   // [B,S]
                     const int*   __restrict__ samples,  // [B,S*15]
                     const float* __restrict__ embed,    // [V,D]
                     float*       __restrict__ out)      // [B,S,S]
{
    const int bid = blockIdx.x;
    const int jt  = bid & 15;           // 16-wide j tile
    const int it  = (bid >> 4) & 7;     // 32-wide i tile
    const int b   = bid >> 7;           // batch

    const int tid    = threadIdx.x;
    const int wave   = tid >> 5;        // 0..7
    const int lane   = tid & 31;
    const int lane16 = lane & 15;
    const int hi     = lane >> 4;       // half-wave id
    const int doff   = hi * 2;          // K-chunk element offset (ISA 32-bit A/B layout)

    // ---- A fragments: 32 latent rows, cached in registers ----
    // lane<16: (d+0,d+1) ; lane>=16: (d+2,d+3)  -> one b64 load per D-chunk
    const float* arow0 = latent + ((size_t)(b * S_ + it * 32 + lane16)) * D_ + doff;
    const float* arow1 = arow0 + 16 * D_;
    v2f afrag0[32], afrag1[32];
    #pragma unroll
    for (int c = 0; c < 32; ++c) {
        afrag0[c] = *(const v2f*)(arow0 + 4 * c);
        afrag1[c] = *(const v2f*)(arow1 + 4 * c);
    }

    // Each wave handles two j's: jt*16 + wave and jt*16 + wave + 8
    #pragma unroll
    for (int jj = 0; jj < 2; ++jj) {
        const int j = jt * 16 + wave + jj * 8;

        // Candidate index for this lane's column n = lane16:
        // n < 15 -> negative sample, n == 15 -> the positive (target label)
        int idx = (lane16 < NNEG)
                    ? samples[(size_t)b * (S_ * NNEG) + j * NNEG + lane16]
                    : labels[b * S_ + j];
        const float* brow = embed + (size_t)idx * D_ + doff;

        v8f acc0 = {}, acc1 = {};
        #pragma unroll
        for (int c = 0; c < 32; ++c) {
            v2f bfrag = *(const v2f*)(brow + 4 * c);   // shared by both WMMAs
            acc0 = __builtin_amdgcn_wmma_f32_16x16x4_f32(
                       false, afrag0[c], false, bfrag, (short)0, acc0, false, false);
            acc1 = __builtin_amdgcn_wmma_f32_16x16x4_f32(
                       false, afrag1[c], false, bfrag, (short)0, acc1, false, false);
        }

        // ---- per-row logsumexp over the 16 candidate columns ----
        // C/D layout: VGPR r, lane L: M = r + (L<16 ? 0 : 8), N = L%16.
        #pragma unroll
        for (int t = 0; t < 2; ++t) {
            v8f acc = t ? acc1 : acc0;
            #pragma unroll
            for (int r = 0; r < 8; ++r) {
                float v = acc[r];
                float m = v;
                #pragma unroll
                for (int off = 1; off < 16; off <<= 1)
                    m = fmaxf(m, __shfl_xor(m, off, 32));
                float e = __expf(v - m);
                #pragma unroll
                for (int off = 1; off < 16; off <<= 1)
                    e += __shfl_xor(e, off, 32);
                // positive score lives at N = 15 of this half-wave
                float pos = __shfl(v, (lane & 16) | 15, 32);
                float res = pos - (m + __logf(e));
                if (lane16 == 0) {
                    int i = it * 32 + t * 16 + r + hi * 8;
                    out[((size_t)(b * S_ + i)) * S_ + j] = res;
                }
            }
        }
    }
}

extern "C" void kernel_launch(void* const* d_in, const int* in_sizes, int n_in,
                              void* d_out, int out_size, void* d_ws, size_t ws_size,
                              hipStream_t stream) {
    (void)in_sizes; (void)n_in; (void)out_size; (void)d_ws; (void)ws_size;
    const float* latent  = (const float*)d_in[0];
    const int*   labels  = (const int*)d_in[1];
    const int*   samples = (const int*)d_in[2];
    const float* embed   = (const float*)d_in[3];
    float*       out     = (float*)d_out;

    dim3 grid(B_ * (S_ / 32) * (S_ / 16));   // 4096 blocks
    neg_loss_kernel<<<grid, 256, 0, stream>>>(latent, labels, samples, embed, out);
}